// MANTIS_PRO_12515534700719
// MI455X (gfx1250) — compile-verified
//
#include <hip/hip_runtime.h>
#include <hip/hip_bf16.h>

typedef __attribute__((ext_vector_type(2))) float v2f;
typedef __attribute__((ext_vector_type(8))) float v8f;

#define B_DIM 128
#define L_DIM 16
#define K_DIM 128   // M*P = 32*4
#define PI_F 3.14159265358979323846f

// One workgroup (4 waves, 128 threads) per batch element b.
// Phase A (all threads): u[l,k] = theta[l,k] + (pi/2^(p+1)) * in[b,l]; cos/sin -> LDS.
// Phase B (all threads): amp = sum_k coef[k] * prod_l cos(u[l,k]);  prob = amp^2.
// Phase C (per wave): normalization = c^T ( Hadamard_l [cos_l cos_l^T + sin_l sin_l^T] ) c.
//   G is symmetric -> only 36 upper-triangular 16x16 tiles (diag w=1, off-diag w=2),
//   built with V_WMMA_F32_16X16X4_F32 (K=2 used, K=2..3 zeroed). Rows paired
//   (0,7)(1,6)(2,5)(3,4) -> exactly 9 tiles per wave, A-operand hoisted per row.
__global__ void __launch_bounds__(128)
mantis_qprob_main(const float* __restrict__ input_ds,
                  const float* __restrict__ theta,
                  const float* __restrict__ coef,
                  float* __restrict__ ws_per_b) {
    __shared__ float sCos[L_DIM * K_DIM];
    __shared__ float sSin[L_DIM * K_DIM];
    __shared__ float sCoef[K_DIM];
    __shared__ float sIn[L_DIM];
    __shared__ float sRed[K_DIM];          // amp reduction
    __shared__ float sNorm[4];             // per-wave normalization partials

    const int tid  = threadIdx.x;          // 0..127
    const int wave = tid >> 5;             // 0..3
    const int lane = tid & 31;             // 0..31 (wave32)
    const int b    = blockIdx.x;           // 0..127

    if (tid < K_DIM) sCoef[tid] = coef[tid];
    if (tid < L_DIM) sIn[tid]   = input_ds[b * L_DIM + tid];
    __syncthreads();

    // ---- Phase A: cos/sin tables (cooperative) -----------------------
    for (int idx = tid; idx < L_DIM * K_DIM; idx += 128) {
        const int l = idx >> 7;            // /128
        const int k = idx & 127;
        const int p = k & 3;               // theta flat = l*128 + m*4 + p
        const float sf = __builtin_amdgcn_ldexpf(PI_F, -(p + 1)); // pi/2^(p+1)
        const float u  = theta[idx] + sf * sIn[l];
        sCos[idx] = __cosf(u);
        sSin[idx] = __sinf(u);
    }
    __syncthreads();

    // ---- Phase B: amplitude (one k-column per thread) ----------------
    {
        float pr = sCos[tid];
#pragma unroll
        for (int l = 1; l < L_DIM; ++l) pr *= sCos[l * K_DIM + tid];
        sRed[tid] = pr * sCoef[tid];
    }
    __syncthreads();
#pragma unroll
    for (int off = 64; off > 0; off >>= 1) {
        if (tid < off) sRed[tid] += sRed[tid + off];
        __syncthreads();
    }

    // ---- Phase C: normalization via symmetric WMMA tiles -------------
    const bool lo = (lane < 16);
    const int  li = lane & 15;
    const int  rowSel = lo ? 0 : 8;        // D layout: VGPR r -> M=r (lo) / r+8 (hi)

    float norm_acc = 0.0f;
    for (int rr = 0; rr < 2; ++rr) {
        const int ti = rr ? (7 - wave) : wave;   // row pairs: 9 tiles per wave
        const int ib = ti * 16;

        // A operands for all l: lanes 0-15 -> (M=lane, K=0/1)=(cos,sin); lanes 16-31 (K=2/3)=0
        v2f A[L_DIM];
#pragma unroll
        for (int l = 0; l < L_DIM; ++l) {
            const float c = sCos[l * K_DIM + ib + li];
            const float s = sSin[l * K_DIM + ib + li];
            A[l].x = lo ? c : 0.0f;
            A[l].y = lo ? s : 0.0f;
        }
        float wrow[8];
#pragma unroll
        for (int r = 0; r < 8; ++r) wrow[r] = sCoef[ib + r + rowSel];

        // diagonal tile (tj == ti): B operand == A operand, weight 1
        {
            v8f z0 = {};
            v8f g = __builtin_amdgcn_wmma_f32_16x16x4_f32(
                        false, A[0], false, A[0], (short)0, z0, false, false);
#pragma unroll
            for (int l = 1; l < L_DIM; ++l) {
                v8f z = {};
                v8f d = __builtin_amdgcn_wmma_f32_16x16x4_f32(
                            false, A[l], false, A[l], (short)0, z, false, false);
#pragma unroll
                for (int r = 0; r < 8; ++r) g[r] *= d[r];
            }
            float t = 0.0f;
#pragma unroll
            for (int r = 0; r < 8; ++r) t += g[r] * wrow[r];
            norm_acc += t * sCoef[ib + li];
        }

        // off-diagonal tiles (tj > ti): weight 2 (symmetry)
        for (int tj = ti + 1; tj < 8; ++tj) {
            const int jb = tj * 16;
            const float wcol = sCoef[jb + li];

            v8f g;
            {
                const float c = sCos[jb + li], s = sSin[jb + li];
                v2f Bv; Bv.x = lo ? c : 0.0f; Bv.y = lo ? s : 0.0f;
                v8f z = {};
                g = __builtin_amdgcn_wmma_f32_16x16x4_f32(
                        false, A[0], false, Bv, (short)0, z, false, false);
            }
#pragma unroll
            for (int l = 1; l < L_DIM; ++l) {
                const float c = sCos[l * K_DIM + jb + li];
                const float s = sSin[l * K_DIM + jb + li];
                v2f Bv; Bv.x = lo ? c : 0.0f; Bv.y = lo ? s : 0.0f;
                v8f z = {};
                v8f d = __builtin_amdgcn_wmma_f32_16x16x4_f32(
                            false, A[l], false, Bv, (short)0, z, false, false);
#pragma unroll
                for (int r = 0; r < 8; ++r) g[r] *= d[r];   // Hadamard over l
            }
            float t = 0.0f;
#pragma unroll
            for (int r = 0; r < 8; ++r) t += g[r] * wrow[r];
            norm_acc += 2.0f * t * wcol;
        }
    }
#pragma unroll
    for (int off = 16; off > 0; off >>= 1) norm_acc += __shfl_xor(norm_acc, off, 32);
    if (lane == 0) sNorm[wave] = norm_acc;
    __syncthreads();

    if (tid == 0) {
        const float amp  = sRed[0];
        const float prob = amp * amp;
        const float norm = sNorm[0] + sNorm[1] + sNorm[2] + sNorm[3];
        ws_per_b[b] = -__logf(prob / norm + 1e-20f);
    }
}

// Single small block: ddof=1 variances + final combine.
__global__ void __launch_bounds__(64)
mantis_qprob_tail(const float* __restrict__ theta,
                  const float* __restrict__ coef,
                  const float* __restrict__ ws_per_b,
                  float* __restrict__ out) {
    __shared__ float redA[64];
    __shared__ float redB[64];
    const int t = threadIdx.x;

    // var over M (32 elems) for group (l,p) = t  -> 64 groups
    float s = 0.0f, s2 = 0.0f;
    {
        const int l = t >> 2, p = t & 3;
        const float* base = theta + l * 128 + p;
        for (int m = 0; m < 32; ++m) { const float v = base[m * 4]; s += v; s2 += v * v; }
    }
    const float var_m = (s2 - s * s * (1.0f / 32.0f)) * (1.0f / 31.0f);

    // var over P (4 elems) for 512 groups (l,m); 8 groups per thread
    float vp = 0.0f;
    for (int g = t; g < 512; g += 64) {
        const float* base = theta + g * 4;
        const float a = base[0], b = base[1], c = base[2], d = base[3];
        const float m4 = 0.25f * (a + b + c + d);
        vp += ((a - m4) * (a - m4) + (b - m4) * (b - m4) +
               (c - m4) * (c - m4) + (d - m4) * (d - m4)) * (1.0f / 3.0f);
    }

    redA[t] = var_m;
    redB[t] = vp;
    __syncthreads();
#pragma unroll
    for (int off = 32; off > 0; off >>= 1) {
        if (t < off) { redA[t] += redA[t + off]; redB[t] += redB[t + off]; }
        __syncthreads();
    }

    if (t == 0) {
        const float reg_m = redA[0] * (1.0f / 64.0f);
        const float reg_p = redB[0] * (1.0f / 512.0f);
        float sc = 0.0f, sc2 = 0.0f;
        for (int k = 0; k < 128; ++k) { const float v = coef[k]; sc += v; sc2 += v * v; }
        const float reg_c = (sc2 - sc * sc * (1.0f / 128.0f)) * (1.0f / 127.0f);
        float acc = 0.0f;
        for (int b = 0; b < B_DIM; ++b) acc += ws_per_b[b];
        out[0] = acc * (1.0f / B_DIM) + 0.01f * (reg_c + reg_m + reg_p);
    }
}

extern "C" void kernel_launch(void* const* d_in, const int* in_sizes, int n_in,
                              void* d_out, int out_size, void* d_ws, size_t ws_size,
                              hipStream_t stream) {
    (void)in_sizes; (void)n_in; (void)out_size; (void)ws_size;
    const float* input_ds = (const float*)d_in[0];  // (128,16)
    const float* theta    = (const float*)d_in[1];  // (16,32,4)
    const float* coef     = (const float*)d_in[2];  // (32,4)
    float*       out      = (float*)d_out;          // scalar
    float*       wsf      = (float*)d_ws;           // >= 128 floats

    mantis_qprob_main<<<dim3(B_DIM), dim3(128), 0, stream>>>(input_ds, theta, coef, wsf);
    mantis_qprob_tail<<<dim3(1), dim3(64), 0, stream>>>(theta, coef, wsf, out);
}